// GateLogisticThresholdExactK_979252543918
// MI455X (gfx1250) — compile-verified
//
#include <hip/hip_runtime.h>
#include <cstdint>

// ---------------------------------------------------------------------------
// Soft top-k gate (logistic threshold, exact-k Newton) for MI455X / gfx1250.
//
// One wave32 per row: row (1024 f32) staged global->LDS with CDNA5 async
// loads, then held in 32 VGPRs/lane as u = exp(s/tau). Newton runs entirely
// in registers: sigmoid((s-t)/tau) = u/(u+v), v = exp(t/tau), one v_rcp_f32
// per element per iteration (no per-element exp inside the loop).
// k-th largest init = 32-step binary search on float bit patterns (exact).
// ---------------------------------------------------------------------------

#define R_DIM          1024
#define ROWS_PER_BLOCK 8
#define WAVE           32
#define ITERS_N        30
#define TAU_F          0.5f
#define LOG2E_F        1.4426950408889634f

__device__ __forceinline__ float wred_f(float x) {
#pragma unroll
  for (int m = 16; m >= 1; m >>= 1) x += __shfl_xor(x, m, 32);
  return x;
}
__device__ __forceinline__ int wred_i(int x) {
#pragma unroll
  for (int m = 16; m >= 1; m >>= 1) x += __shfl_xor(x, m, 32);
  return x;
}

__global__ __launch_bounds__(ROWS_PER_BLOCK * WAVE)
void gate_topk_newton_kernel(const float* __restrict__ s,
                             const unsigned char* __restrict__ mask,
                             const int* __restrict__ kptr,
                             float* __restrict__ out, int B)
{
  __shared__ __align__(16) float smem[ROWS_PER_BLOCK][R_DIM];

  const int lane = threadIdx.x & (WAVE - 1);
  const int wv   = threadIdx.x >> 5;
  const int row  = blockIdx.x * ROWS_PER_BLOCK + wv;
  if (row >= B) return;                       // uniform per wave

  const int   k_eff   = min(kptr[0], R_DIM);
  const float kf      = (float)k_eff;
  const float inv_tau = 1.0f / TAU_F;         // tau = max(0.5, 1e-6)
  const float cexp    = inv_tau * LOG2E_F;    // exp(x/tau) = exp2(x*cexp)

  // ---- CDNA5 async stage: row -> LDS (8 x b128 per lane = 4 KB/row) ----
  const uint32_t lds0 = (uint32_t)(uintptr_t)(&smem[wv][0]);
  const uint64_t gb   = (uint64_t)(uintptr_t)s;
#pragma unroll
  for (int i = 0; i < 8; ++i) {
    uint32_t goff = (uint32_t)row * (R_DIM * 4) + i * 512 + lane * 16;
    uint32_t loff = lds0 + i * 512 + lane * 16;
    asm volatile("global_load_async_to_lds_b128 %0, %1, %2"
                 :: "v"(loff), "v"(goff), "s"(gb) : "memory");
  }

  // mask bytes loaded directly (4 bytes/lane per chunk, coalesced)
  uint32_t mw[8];
  const uint32_t* m32 = (const uint32_t*)mask;
#pragma unroll
  for (int i = 0; i < 8; ++i)
    mw[i] = m32[((size_t)row * R_DIM + (size_t)(i * 128 + lane * 4)) >> 2];

  asm volatile("s_wait_asynccnt 0" ::: "memory");

  // ---- u = mask ? exp(s/tau) : 0   (masked == exp(BIG_NEG/tau) == 0) ----
  float u[32];
#pragma unroll
  for (int i = 0; i < 8; ++i) {
    const float4 q = *(const float4*)(&smem[wv][i * 128 + lane * 4]);
    const float e0 = __builtin_amdgcn_exp2f(q.x * cexp);
    const float e1 = __builtin_amdgcn_exp2f(q.y * cexp);
    const float e2 = __builtin_amdgcn_exp2f(q.z * cexp);
    const float e3 = __builtin_amdgcn_exp2f(q.w * cexp);
    u[i * 4 + 0] = (mw[i] & 0x000000ffu) ? e0 : 0.0f;
    u[i * 4 + 1] = (mw[i] & 0x0000ff00u) ? e1 : 0.0f;
    u[i * 4 + 2] = (mw[i] & 0x00ff0000u) ? e2 : 0.0f;
    u[i * 4 + 3] = (mw[i] & 0xff000000u) ? e3 : 0.0f;
  }

  // ---- exact k-th largest of u via binary search on bit patterns ----
  // u >= 0, so uint bit order == float order. Finds max x with count(u>=x)>=k,
  // which is exactly the k-th largest value (== exp(t0/tau) of reference).
  uint32_t x = 0u;
#pragma unroll 1
  for (int b = 31; b >= 0; --b) {
    const uint32_t cand = x | (1u << b);
    const float    cf   = __uint_as_float(cand);
    int cnt = 0;
#pragma unroll
    for (int j = 0; j < 32; ++j) cnt += (u[j] >= cf) ? 1 : 0;
    cnt = wred_i(cnt);
    if (cnt >= k_eff) x = cand;
  }
  float v = __uint_as_float(x);   // v = exp(t/tau)

  // ---- 30 Newton iterations, all in registers ----
#pragma unroll 1
  for (int it = 0; it < ITERS_N; ++it) {
    float sg = 0.0f, sgp = 0.0f;
#pragma unroll
    for (int j = 0; j < 32; ++j) {
      const float uu  = u[j];
      const float r   = __builtin_amdgcn_rcpf(uu + v); // v_rcp_f32
      const float g   = uu * r;                        // sigmoid((s-t)/tau)
      const float omg = v  * r;                        // 1 - g
      sg  += g;
      sgp  = __builtin_fmaf(g, omg, sgp);              // g*(1-g)
    }
    sg  = wred_f(sg);
    sgp = wred_f(sgp);
    const float Fk    = sg - kf;
    const float dF    = -sgp * inv_tau;
    const float delta = Fk * __builtin_amdgcn_rcpf(dF + 1e-8f);
    // t -= delta  =>  v *= exp(-delta/tau)   (1 exp per row per iter)
    v = v * __builtin_amdgcn_exp2f(-delta * cexp);
  }

  // ---- final gate, clip, coalesced b128 stores ----
#pragma unroll
  for (int i = 0; i < 8; ++i) {
    float4 o;
    const float r0 = __builtin_amdgcn_rcpf(u[i*4+0] + v);
    const float r1 = __builtin_amdgcn_rcpf(u[i*4+1] + v);
    const float r2 = __builtin_amdgcn_rcpf(u[i*4+2] + v);
    const float r3 = __builtin_amdgcn_rcpf(u[i*4+3] + v);
    o.x = fminf(fmaxf(u[i*4+0] * r0, 0.0f), 1.0f);
    o.y = fminf(fmaxf(u[i*4+1] * r1, 0.0f), 1.0f);
    o.z = fminf(fmaxf(u[i*4+2] * r2, 0.0f), 1.0f);
    o.w = fminf(fmaxf(u[i*4+3] * r3, 0.0f), 1.0f);
    *(float4*)(&out[(size_t)row * R_DIM + (size_t)(i * 128 + lane * 4)]) = o;
  }
}

extern "C" void kernel_launch(void* const* d_in, const int* in_sizes, int n_in,
                              void* d_out, int out_size, void* d_ws, size_t ws_size,
                              hipStream_t stream) {
  const float*         sArr = (const float*)d_in[0];
  const unsigned char* mArr = (const unsigned char*)d_in[1];   // jnp bool = 1 byte
  const int*           kPtr = (const int*)d_in[2];
  float*               oArr = (float*)d_out;

  const int B      = in_sizes[0] / R_DIM;                      // 16384
  const int blocks = (B + ROWS_PER_BLOCK - 1) / ROWS_PER_BLOCK;

  gate_topk_newton_kernel<<<blocks, ROWS_PER_BLOCK * WAVE, 0, stream>>>(
      sArr, mArr, kPtr, oArr, B);
}